// GCNNet_48318382080353
// MI455X (gfx1250) — compile-verified
//
#include <hip/hip_runtime.h>
#include <hip/hip_bf16.h>

typedef __attribute__((ext_vector_type(2))) float v2f;
typedef __attribute__((ext_vector_type(8))) float v8f;

// ---------------------------------------------------------------------------
// Workspace layout (floats), Npad = N rounded up to 64:
//   dinv : [Npad]        (starts as degree accumulator)
//   tsum : [Npad]        scalar layer-1 edge accumulator
//   s    : [Npad]        per-node scalar after layer-1 GCN
//   h2   : [Npad*64]     relu(s*W1+b1) @ W2
//   z    : [Npad*64]     layer-2 GCN output
// ---------------------------------------------------------------------------

__global__ __launch_bounds__(256) void k_init(float* deg, float* tsum, int N) {
    int i = blockIdx.x * blockDim.x + threadIdx.x;
    if (i < N) { deg[i] = 1.0f; tsum[i] = 0.0f; }  // 1.0 = self loop
}

__global__ __launch_bounds__(256) void k_count(const int* __restrict__ dst,
                                               float* __restrict__ deg, int E) {
    int e = blockIdx.x * blockDim.x + threadIdx.x;
    if (e < E) atomicAdd(&deg[dst[e]], 1.0f);
}

__global__ __launch_bounds__(256) void k_dinv(float* d, int N) {
    int i = blockIdx.x * blockDim.x + threadIdx.x;
    if (i < N) d[i] = rsqrtf(d[i]);
}

// tsum[dst] += x[src] * dinv[src]   (scalar layer-1 message aggregation)
__global__ __launch_bounds__(256) void k_agg1(const int* __restrict__ src,
                                              const int* __restrict__ dst,
                                              const float* __restrict__ x,
                                              const float* __restrict__ dinv,
                                              float* __restrict__ tsum, int E) {
    int e = blockIdx.x * blockDim.x + threadIdx.x;
    if (e < E) {
        int si = src[e], di = dst[e];
        atomicAdd(&tsum[di], x[si] * dinv[si]);
    }
}

// s[i] = dinv[i]*tsum[i] + x[i]*dinv[i]^2
__global__ __launch_bounds__(256) void k_finish_s(const float* __restrict__ x,
                                                  const float* __restrict__ dinv,
                                                  const float* __restrict__ tsum,
                                                  float* __restrict__ s, int N) {
    int i = blockIdx.x * blockDim.x + threadIdx.x;
    if (i < N) {
        float dv = dinv[i];
        s[i] = dv * tsum[i] + x[i] * dv * dv;
    }
}

// ---------------------------------------------------------------------------
// H2[N,64] = relu(s[:,None]*W1[None,:] + b1) @ W2      (fp32 WMMA 16x16x4)
// 256 threads = 8 waves; each wave owns a 16-row tile, 4 accumulators span
// the 64 output columns. A fragment is synthesized on the fly from s/W1/b1.
// A layout (16x4 f32): lanes 0-15 -> (M=lane, K={k,k+1}); lanes 16-31 ->
// (M=lane-16, K={k+2,k+3}).  B layout mirrors with N=lane%16.
// C/D layout: reg r -> row r (lanes 0-15) / row r+8 (lanes 16-31), col lane%16.
// ---------------------------------------------------------------------------
__global__ __launch_bounds__(256) void k_gemm_wmma(const float* __restrict__ s,
                                                   const float* __restrict__ W1g,
                                                   const float* __restrict__ b1g,
                                                   const float* __restrict__ W2g,
                                                   float* __restrict__ h2, int N) {
    __shared__ float sW2[128 * 64];
    __shared__ float sW1[128];
    __shared__ float sB1[128];
    int tid = threadIdx.x;
    for (int idx = tid; idx < 128 * 64; idx += 256) sW2[idx] = W2g[idx];
    if (tid < 128) { sW1[tid] = W1g[tid]; sB1[tid] = b1g[tid]; }
    __syncthreads();

    const int wave = tid >> 5;
    const int lane = tid & 31;
    const int mrow = lane & 15;
    const int khalf = (lane >> 4) << 1;      // 0 for lanes 0-15, 2 for 16-31
    const int mbase = blockIdx.x * 128 + wave * 16;
    const int gm = mbase + mrow;
    const float sv = (gm < N) ? s[gm] : 0.0f;

    v8f acc0 = {}, acc1 = {}, acc2 = {}, acc3 = {};

    for (int k = 0; k < 128; k += 4) {
        const int k0 = k + khalf;
        // A fragment: relu(s*W1 + b1) for (row=mrow, K=k0..k0+1)
        float a0 = fmaf(sv, sW1[k0], sB1[k0]);
        float a1 = fmaf(sv, sW1[k0 + 1], sB1[k0 + 1]);
        v2f a;
        a.x = a0 > 0.0f ? a0 : 0.0f;
        a.y = a1 > 0.0f ? a1 : 0.0f;
        // B fragments for the 4 column blocks
        const float* r0 = &sW2[k0 * 64];
        const float* r1 = &sW2[(k0 + 1) * 64];
        const int n = mrow;
        v2f b0, b1, b2, b3;
        b0.x = r0[n];      b0.y = r1[n];
        b1.x = r0[n + 16]; b1.y = r1[n + 16];
        b2.x = r0[n + 32]; b2.y = r1[n + 32];
        b3.x = r0[n + 48]; b3.y = r1[n + 48];

        acc0 = __builtin_amdgcn_wmma_f32_16x16x4_f32(false, a, false, b0, (short)0, acc0, false, false);
        acc1 = __builtin_amdgcn_wmma_f32_16x16x4_f32(false, a, false, b1, (short)0, acc1, false, false);
        acc2 = __builtin_amdgcn_wmma_f32_16x16x4_f32(false, a, false, b2, (short)0, acc2, false, false);
        acc3 = __builtin_amdgcn_wmma_f32_16x16x4_f32(false, a, false, b3, (short)0, acc3, false, false);
    }

    const int rowoff = (lane >> 4) * 8;
    const int col = lane & 15;
#pragma unroll
    for (int r = 0; r < 8; ++r) {
        int row = mbase + rowoff + r;
        if (row < N) {
            float* o = h2 + (size_t)row * 64 + col;
            o[0]  = acc0[r];
            o[16] = acc1[r];
            o[32] = acc2[r];
            o[48] = acc3[r];
        }
    }
}

// z[i,f] = h2[i,f]*dinv[i]^2 + b2[f]   (self-loop term + bias, init for atomics)
__global__ __launch_bounds__(256) void k_init_z(const float* __restrict__ h2,
                                                const float* __restrict__ dinv,
                                                const float* __restrict__ b2,
                                                float* __restrict__ z, int N) {
    int idx = blockIdx.x * blockDim.x + threadIdx.x;
    if (idx < N * 64) {
        int i = idx >> 6;
        int f = idx & 63;
        float dv = dinv[i];
        z[idx] = h2[idx] * dv * dv + b2[f];
    }
}

// z[dst,:] += h2[src,:] * dinv[src]*dinv[dst]    (4 edges per 256-thread block)
__global__ __launch_bounds__(256) void k_agg2(const int* __restrict__ src,
                                              const int* __restrict__ dst,
                                              const float* __restrict__ dinv,
                                              const float* __restrict__ h2,
                                              float* __restrict__ z, int E) {
    int e = blockIdx.x * 4 + (threadIdx.x >> 6);
    int f = threadIdx.x & 63;
    if (e >= E) return;
    int si = src[e], di = dst[e];
    float nrm = dinv[si] * dinv[di];
    atomicAdd(&z[(size_t)di * 64 + f], h2[(size_t)si * 64 + f] * nrm);
}

// logits[k] = dot(z[i], z[j]), one wave32 per edge, float2 per lane.
__global__ __launch_bounds__(256) void k_decode(const int* __restrict__ ei,
                                                const int* __restrict__ nei,
                                                const float* __restrict__ z,
                                                float* __restrict__ out, int E) {
    int lane = threadIdx.x & 31;
    int k = blockIdx.x * 8 + (threadIdx.x >> 5);
    if (k >= 2 * E) return;
    int i, j;
    if (k < E) { i = ei[k];        j = ei[E + k]; }
    else       { i = nei[k - E];   j = nei[E + (k - E)]; }
    const float2* za = (const float2*)(z + (size_t)i * 64);
    const float2* zb = (const float2*)(z + (size_t)j * 64);
    float2 a = za[lane];
    float2 b = zb[lane];
    float p = a.x * b.x + a.y * b.y;
#pragma unroll
    for (int m = 16; m >= 1; m >>= 1) p += __shfl_xor(p, m, 32);
    if (lane == 0) out[k] = p;
}

extern "C" void kernel_launch(void* const* d_in, const int* in_sizes, int n_in,
                              void* d_out, int out_size, void* d_ws, size_t ws_size,
                              hipStream_t stream) {
    const float* x   = (const float*)d_in[0];
    const int*   ei  = (const int*)d_in[1];   // [2,E] row-major: src then dst
    const int*   nei = (const int*)d_in[2];
    const float* W1  = (const float*)d_in[3];
    const float* b1  = (const float*)d_in[4];
    const float* W2  = (const float*)d_in[5];
    const float* b2  = (const float*)d_in[6];
    float* out = (float*)d_out;

    const int N = in_sizes[0];          // 100000 (x is [N,1])
    const int E = in_sizes[1] / 2;      // 1600000
    const int Npad = (N + 63) & ~63;

    float* ws   = (float*)d_ws;
    float* dinv = ws;                               // [Npad]  (degree, then rsqrt)
    float* tsum = ws + Npad;                        // [Npad]
    float* s    = ws + 2 * (size_t)Npad;            // [Npad]
    float* h2   = ws + 3 * (size_t)Npad;            // [Npad*64]
    float* z    = h2 + (size_t)Npad * 64;           // [Npad*64]

    const int* src = ei;
    const int* dst = ei + E;

    const int T = 256;
    k_init   <<<(N + T - 1) / T, T, 0, stream>>>(dinv, tsum, N);
    k_count  <<<(E + T - 1) / T, T, 0, stream>>>(dst, dinv, E);
    k_dinv   <<<(N + T - 1) / T, T, 0, stream>>>(dinv, N);
    k_agg1   <<<(E + T - 1) / T, T, 0, stream>>>(src, dst, x, dinv, tsum, E);
    k_finish_s<<<(N + T - 1) / T, T, 0, stream>>>(x, dinv, tsum, s, N);
    k_gemm_wmma<<<(N + 127) / 128, T, 0, stream>>>(s, W1, b1, W2, h2, N);
    k_init_z <<<(N * 64 + T - 1) / T, T, 0, stream>>>(h2, dinv, b2, z, N);
    k_agg2   <<<(E + 3) / 4, T, 0, stream>>>(src, dst, dinv, h2, z, E);
    k_decode <<<(2 * E + 7) / 8, T, 0, stream>>>(ei, nei, z, out, E);
}